// SelectiveSSM_16544214024237
// MI455X (gfx1250) — compile-verified
//
#include <hip/hip_runtime.h>
#include <stdint.h>

// Problem constants (match the reference)
#define D_MODEL_ 1024
#define D_STATE_ 16
#define SEQ_     2048
#define BATCH_   2

// Tiling
#define D_TILE   8                 // d-channels per block (1 wave handles 2)
#define T_CHUNK  128               // timesteps staged per TDM chunk
#define NCHUNK   (SEQ_ / T_CHUNK)  // 16
#define NTHREADS 128               // 4 waves (wave32)

typedef uint32_t u32x4 __attribute__((ext_vector_type(4)));
typedef int      i32x8 __attribute__((ext_vector_type(8)));
typedef int      i32x4 __attribute__((ext_vector_type(4)));

// Build a 2-D tile D# and issue TENSOR_LOAD_TO_LDS (TENSORcnt-tracked DMA).
// tile0 elements contiguous (dim0), tile1 rows with stride0 elements between
// rows, fp32 data. Groups 2/3/4 zero: tile_dim2 == 0 -> dims >= 2 unused.
static __device__ __forceinline__ void tdm_load_2d(
    uint32_t lds_byte_off, const void* gaddr,
    uint32_t tile0, uint32_t tile1, uint64_t stride0_elems)
{
  uint64_t ga = (uint64_t)(uintptr_t)gaddr;
  u32x4 g0;
  g0.x = 1u;                                   // count=1 (valid user D#)
  g0.y = lds_byte_off;                         // LDS destination byte address
  g0.z = (uint32_t)ga;                         // global_addr[31:0]
  g0.w = (uint32_t)(ga >> 32) | (2u << 30);    // global_addr[56:32] | type=2
  i32x8 g1;
  g1[0] = (int)(2u << 16);                     // workgroup_mask=0, data_size=2 (4B)
  g1[1] = (int)((tile0 & 0xffffu) << 16);                   // tensor_dim0[15:0]
  g1[2] = (int)((tile0 >> 16) | ((tile1 & 0xffffu) << 16)); // dim0 hi | tensor_dim1 lo
  g1[3] = (int)((tile1 >> 16) | ((tile0 & 0xffffu) << 16)); // dim1 hi | tile_dim0
  g1[4] = (int)(tile1 & 0xffffu);                           // tile_dim1 (tile_dim2=0)
  g1[5] = (int)(uint32_t)(stride0_elems & 0xffffffffull);   // tensor_dim0_stride lo
  g1[6] = (int)(uint32_t)((stride0_elems >> 32) & 0xffffull);
  g1[7] = 0;
  i32x4 z4 = {0, 0, 0, 0};
  i32x8 z8 = {0, 0, 0, 0, 0, 0, 0, 0};
  __builtin_amdgcn_tensor_load_to_lds(g0, g1, z4, z4, z8, 0);
}

__global__ __launch_bounds__(NTHREADS)
void ssm_scan_kernel(const float* __restrict__ x,   const float* __restrict__ dt,
                     const float* __restrict__ Bm,  const float* __restrict__ Cm,
                     const float* __restrict__ Alog,const float* __restrict__ Dv,
                     float* __restrict__ out)
{
  __shared__ float s_dt[2][T_CHUNK * D_TILE];   //  8 KB
  __shared__ float s_x [2][T_CHUNK * D_TILE];   //  8 KB
  __shared__ float s_B [2][T_CHUNK * D_STATE_]; // 16 KB
  __shared__ float s_C [2][T_CHUNK * D_STATE_]; // 16 KB
  __shared__ float s_y [T_CHUNK * D_TILE];      //  4 KB

  const int tid  = threadIdx.x;
  const int wid  = __builtin_amdgcn_readfirstlane(tid >> 5); // uniform wave id
  const int lane = tid & 31;
  const int half = lane >> 4;       // which of the wave's two d-channels
  const int n    = lane & 15;       // state index
  const int dloc = ((tid >> 5) << 1) + half;   // 0..7 within tile
  const int d0   = blockIdx.x * D_TILE;
  const int b    = blockIdx.y;
  const int d    = d0 + dloc;

  // A[d,n] = -exp(A_log); pre-scale by log2(e) so the recurrence uses a raw
  // v_exp_f32 (exp2) with no extra dependent multiply in the serial chain.
  const float a2 = -__expf(Alog[d * D_STATE_ + n]) * 1.44269504088896340736f;
  const float Dd = Dv[d];

  // Per-wave TDM stream: wave0->dt, wave1->x, wave2->B, wave3->C.
  const float* gp0;  uint32_t tile0;  uint64_t stride0;  size_t gstep;
  const void *lb0, *lb1;
  if (wid == 0) {
    gp0 = dt + ((size_t)b * SEQ_) * D_MODEL_ + d0;
    tile0 = D_TILE; stride0 = D_MODEL_; gstep = (size_t)T_CHUNK * D_MODEL_;
    lb0 = &s_dt[0][0]; lb1 = &s_dt[1][0];
  } else if (wid == 1) {
    gp0 = x + ((size_t)b * SEQ_) * D_MODEL_ + d0;
    tile0 = D_TILE; stride0 = D_MODEL_; gstep = (size_t)T_CHUNK * D_MODEL_;
    lb0 = &s_x[0][0]; lb1 = &s_x[1][0];
  } else if (wid == 2) {
    gp0 = Bm + ((size_t)b * SEQ_) * D_STATE_;
    tile0 = D_STATE_; stride0 = D_STATE_; gstep = (size_t)T_CHUNK * D_STATE_;
    lb0 = &s_B[0][0]; lb1 = &s_B[1][0];
  } else {
    gp0 = Cm + ((size_t)b * SEQ_) * D_STATE_;
    tile0 = D_STATE_; stride0 = D_STATE_; gstep = (size_t)T_CHUNK * D_STATE_;
    lb0 = &s_C[0][0]; lb1 = &s_C[1][0];
  }

  // Prefetch chunk 0
  tdm_load_2d((uint32_t)(uintptr_t)lb0, gp0, tile0, T_CHUNK, stride0);

  float s = 0.0f;   // scan state S[d,n], one element per lane

  for (int c = 0; c < NCHUNK; ++c) {
    // Prefetch chunk c+1 into the other buffer, then wait for chunk c
    // (per-wave TENSORcnt is in-order: <=1 means the older load retired).
    if (c + 1 < NCHUNK) {
      const void* lbn = ((c + 1) & 1) ? lb1 : lb0;
      tdm_load_2d((uint32_t)(uintptr_t)lbn, gp0 + (size_t)(c + 1) * gstep,
                  tile0, T_CHUNK, stride0);
      __builtin_amdgcn_s_wait_tensorcnt((short)1);
    } else {
      __builtin_amdgcn_s_wait_tensorcnt((short)0);
    }
    __syncthreads();   // all 4 waves' tiles for chunk c visible; s_y reusable

    const float* pdt = s_dt[c & 1];
    const float* px  = s_x [c & 1];
    const float* pB  = s_B [c & 1];
    const float* pC  = s_C [c & 1];

    #pragma unroll 4
    for (int t = 0; t < T_CHUNK; ++t) {
      const float dtv = pdt[t * D_TILE   + dloc];  // LDS broadcast (16 lanes)
      const float xv  = px [t * D_TILE   + dloc];
      const float bv  = pB [t * D_STATE_ + n];
      const float cv  = pC [t * D_STATE_ + n];
      const float da  = __builtin_amdgcn_exp2f(dtv * a2); // exp(dt*A[d,n])
      s = __builtin_fmaf(da, s, (dtv * xv) * bv);  // state update
      float p = s * cv;                            // y contribution
      p += __shfl_xor(p, 1, 32);                   // reduce over n (16 lanes)
      p += __shfl_xor(p, 2, 32);
      p += __shfl_xor(p, 4, 32);
      p += __shfl_xor(p, 8, 32);
      if (n == 0)
        s_y[t * D_TILE + dloc] = __builtin_fmaf(xv, Dd, p);  // + x*D skip
    }
    __syncthreads();

    // Coalesced float4 stores of the chunk's y (128 t-rows x 8 d).
    #pragma unroll
    for (int k = 0; k < 2; ++k) {
      const int trow = (tid >> 1) + 64 * k;
      const int part = tid & 1;
      float4 v = *(const float4*)&s_y[trow * D_TILE + part * 4];
      const size_t t_abs = (size_t)c * T_CHUNK + trow;
      *(float4*)&out[((size_t)b * SEQ_ + t_abs) * D_MODEL_ + d0 + part * 4] = v;
    }
    // Next iteration's top __syncthreads fences s_y before it is rewritten.
  }
  __builtin_amdgcn_s_wait_tensorcnt((short)0);
}

extern "C" void kernel_launch(void* const* d_in, const int* in_sizes, int n_in,
                              void* d_out, int out_size, void* d_ws, size_t ws_size,
                              hipStream_t stream) {
  (void)in_sizes; (void)n_in; (void)out_size; (void)d_ws; (void)ws_size;
  const float* x    = (const float*)d_in[0];
  const float* dt   = (const float*)d_in[1];
  const float* Bm   = (const float*)d_in[2];
  const float* Cm   = (const float*)d_in[3];
  const float* Alog = (const float*)d_in[4];
  const float* Dv   = (const float*)d_in[5];
  float* out = (float*)d_out;

  dim3 grid(D_MODEL_ / D_TILE, BATCH_);   // 128 x 2 = 256 blocks
  ssm_scan_kernel<<<grid, NTHREADS, 0, stream>>>(x, dt, Bm, Cm, Alog, Dv, out);
}